// CrossCompress_86655260164823
// MI455X (gfx1250) — compile-verified
//
#include <hip/hip_runtime.h>
#include <hip/hip_bf16.h>

typedef __attribute__((ext_vector_type(2))) float v2f;
typedef __attribute__((ext_vector_type(4))) float v4f;
typedef __attribute__((ext_vector_type(8))) float v8f;

#define DD 128
#define TILE_M 16
#define WAVES_PER_BLOCK 8

// One wave processes a tile of 16 rows.
// Pass 1: C_v = V_tile(16x128) x [w_ev,w_ee,0...](128x16)   via 32x V_WMMA_F32_16X16X4_F32
//         C_e = E_tile(16x128) x [w_vv,w_ve,0...](128x16)
// Pass 2: broadcast per-row dots from C, stream tile again as float4/lane,
//         item_out = v*(e.w_vv) + e*(v.w_ev) + bias_v
//         ent_out  = v*(e.w_ve) + e*(v.w_ee) + bias_e
__global__ __launch_bounds__(256, 1) void CrossCompress_86655260164823_kernel(
    const float* __restrict__ v_in, const float* __restrict__ e_in,
    const float* __restrict__ w_vv, const float* __restrict__ w_ve,
    const float* __restrict__ w_ev, const float* __restrict__ w_ee,
    const float* __restrict__ bias_v, const float* __restrict__ bias_e,
    float* __restrict__ item_out, float* __restrict__ ent_out, int nrows)
{
    __shared__ __align__(16) float lds_w[4][DD]; // 0:w_vv 1:w_ve 2:w_ev 3:w_ee

    const int t = threadIdx.x;
    for (int i = t; i < 4 * DD; i += 256) {
        const int which = i >> 7;
        const int k = i & (DD - 1);
        const float* src = (which == 0) ? w_vv : (which == 1) ? w_ve
                         : (which == 2) ? w_ev : w_ee;
        lds_w[which][k] = src[k];
    }
    __syncthreads();

    const int wave = t >> 5;
    const int lane = t & 31;
    const long tile = (long)blockIdx.x * WAVES_PER_BLOCK + wave;
    const long row0 = tile * TILE_M;
    if (row0 >= nrows) return; // uniform per wave; no barriers after this

    // A-matrix lane mapping for V_WMMA_F32_16X16X4_F32:
    //   lanes 0-15: row m = lane,     holds K = k0, k0+1
    //   lanes 16-31: row m = lane-16, holds K = k0+2, k0+3
    // B-matrix: lane = column N (same K pairing). We use N=0,1 only.
    const int m    = lane & 15;
    const int kSel = (lane >> 4) << 1;

    long arow = row0 + m;
    if (arow > (long)nrows - 1) arow = (long)nrows - 1; // clamp (tail safety)
    const float* vrow = v_in + arow * DD;
    const float* erow = e_in + arow * DD;

    const float bscale = (m < 2) ? 1.0f : 0.0f;          // zero-pad cols 2..15
    const float* wv_col = lds_w[(m == 1) ? 3 : 2];       // B for C_v: [w_ev | w_ee]
    const float* we_col = lds_w[(m == 1) ? 1 : 0];       // B for C_e: [w_vv | w_ve]

    v8f cv = {};
    v8f ce = {};
#pragma unroll 4
    for (int k0 = 0; k0 < DD; k0 += 4) {
        const int k = k0 + kSel;
        v2f av = { vrow[k],   vrow[k + 1] };
        v2f ae = { erow[k],   erow[k + 1] };
        v2f bv = { wv_col[k] * bscale, wv_col[k + 1] * bscale };
        v2f be = { we_col[k] * bscale, we_col[k + 1] * bscale };
        cv = __builtin_amdgcn_wmma_f32_16x16x4_f32(false, av, false, bv,
                                                   (short)0, cv, false, false);
        ce = __builtin_amdgcn_wmma_f32_16x16x4_f32(false, ae, false, be,
                                                   (short)0, ce, false, false);
    }

    // Epilogue: lane handles columns 4*lane .. 4*lane+3 of each row.
    const v4f b_v = *(const v4f*)(bias_v + 4 * lane);
    const v4f b_e = *(const v4f*)(bias_e + 4 * lane);

#pragma unroll
    for (int r = 0; r < TILE_M; ++r) {
        const long row = row0 + r;
        if (row >= nrows) break;
        // C element (r, n): VGPR r&7, lane (r<8 ? 0 : 16) + n
        const int srcLane = (r < 8) ? 0 : 16;
        const float d_evv = __shfl(ce[r & 7], srcLane + 0, 32); // e . w_vv
        const float d_eve = __shfl(ce[r & 7], srcLane + 1, 32); // e . w_ve
        const float d_vev = __shfl(cv[r & 7], srcLane + 0, 32); // v . w_ev
        const float d_vee = __shfl(cv[r & 7], srcLane + 1, 32); // v . w_ee

        const v4f vx = *(const v4f*)(v_in + row * DD + 4 * lane);
        const v4f ex = *(const v4f*)(e_in + row * DD + 4 * lane);

        v4f io, eo;
        io.x = vx.x * d_evv + ex.x * d_vev + b_v.x;
        io.y = vx.y * d_evv + ex.y * d_vev + b_v.y;
        io.z = vx.z * d_evv + ex.z * d_vev + b_v.z;
        io.w = vx.w * d_evv + ex.w * d_vev + b_v.w;
        eo.x = vx.x * d_eve + ex.x * d_vee + b_e.x;
        eo.y = vx.y * d_eve + ex.y * d_vee + b_e.y;
        eo.z = vx.z * d_eve + ex.z * d_vee + b_e.z;
        eo.w = vx.w * d_eve + ex.w * d_vee + b_e.w;

        *(v4f*)(item_out + row * DD + 4 * lane) = io;
        *(v4f*)(ent_out  + row * DD + 4 * lane) = eo;
    }
}

extern "C" void kernel_launch(void* const* d_in, const int* in_sizes, int n_in,
                              void* d_out, int out_size, void* d_ws, size_t ws_size,
                              hipStream_t stream) {
    const float* v_in   = (const float*)d_in[0]; // item_embedding [B,128,1]
    const float* e_in   = (const float*)d_in[1]; // entity_embedding
    const float* w_vv   = (const float*)d_in[2];
    const float* w_ve   = (const float*)d_in[3];
    const float* w_ev   = (const float*)d_in[4];
    const float* w_ee   = (const float*)d_in[5];
    const float* bias_v = (const float*)d_in[6];
    const float* bias_e = (const float*)d_in[7];

    const int nrows = in_sizes[0] / DD; // B = 16384
    float* out      = (float*)d_out;
    float* item_out = out;                          // first output, B*D floats
    float* ent_out  = out + (size_t)nrows * DD;     // second output

    const int tiles = (nrows + TILE_M - 1) / TILE_M;
    const int grid  = (tiles + WAVES_PER_BLOCK - 1) / WAVES_PER_BLOCK;

    hipLaunchKernelGGL(CrossCompress_86655260164823_kernel,
                       dim3(grid), dim3(256), 0, stream,
                       v_in, e_in, w_vv, w_ve, w_ev, w_ee, bias_v, bias_e,
                       item_out, ent_out, nrows);
}